// RationalQuadraticSpline_38379827757688
// MI455X (gfx1250) — compile-verified
//
#include <hip/hip_runtime.h>
#include <math.h>

// ---------------------------------------------------------------------------
// Rational-quadratic spline (neural spline flow), B=500000, V=16, K=30.
// Memory-bound elementwise op: 32MB in, 64MB out => ~4.1us floor @ 23.3TB/s.
// No contraction dim => WMMA not applicable. CDNA5 features used instead:
//   * global_load_async_to_lds_b128 + s_wait_asynccnt (async table staging)
//   * global_prefetch_b8 via __builtin_prefetch on the streaming input
//   * wave32-friendly LDS layout (odd row stride 33 => 16 V-rows hit 16 banks)
// ---------------------------------------------------------------------------

#define RQS_V   16
#define RQS_K   30
#define RQS_STR 33                    // floats per row (odd => bank-conflict-free)
#define RQS_ROW (RQS_V * RQS_STR)     // 528 floats per table
#define OFF_CW  0                     // cumwidths  (K+1 valid)
#define OFF_CH  (1 * RQS_ROW)         // cumheights (K+1 valid)
#define OFF_DV  (2 * RQS_ROW)         // derivatives(K+1 valid)
#define OFF_W   (3 * RQS_ROW)         // widths     (K valid)
#define OFF_H   (4 * RQS_ROW)         // heights    (K valid)
#define OFF_DL  (5 * RQS_ROW)         // delta      (K valid)
#define TBL_FLOATS (6 * RQS_ROW)      // 3168 floats = 12672 bytes
#define TBL_VEC4   (TBL_FLOATS / 4)   // 792 b128 transfers

// ---------------- table-build kernel (tiny, 1 block) -----------------------

__device__ __forceinline__ void build_knots(const float* __restrict__ un,
                                            float* __restrict__ ws,
                                            int v, int off_c, int off_w) {
    const float lo = -5.0f, hi = 5.0f;
    float m = -3.402823466e38f;
    for (int k = 0; k < RQS_K; ++k) m = fmaxf(m, un[v * RQS_K + k]);
    float e[RQS_K];
    float s = 0.0f;
    for (int k = 0; k < RQS_K; ++k) { e[k] = expf(un[v * RQS_K + k] - m); s += e[k]; }
    float inv = 1.0f / s;
    float acc = 0.0f, prev = lo;
    ws[off_c + v * RQS_STR] = lo;
    for (int k = 0; k < RQS_K; ++k) {
        float w = 1e-3f + (1.0f - 1e-3f * (float)RQS_K) * (e[k] * inv);
        acc += w;
        float c = (k == RQS_K - 1) ? hi : ((hi - lo) * acc + lo);
        ws[off_c + v * RQS_STR + k + 1] = c;
        ws[off_w + v * RQS_STR + k]     = c - prev;   // widths from final knots
        prev = c;
    }
}

__global__ void rqs_tables_kernel(const float* __restrict__ uw,
                                  const float* __restrict__ uh,
                                  const float* __restrict__ ud,
                                  float* __restrict__ ws) {
    int v = threadIdx.x;
    if (v >= RQS_V) return;
    build_knots(uw, ws, v, OFF_CW, OFF_W);
    build_knots(uh, ws, v, OFF_CH, OFF_H);
    // boundary const so that min_deriv + softplus(const) == 1
    const float cst = (float)(log(exp(1.0 - 1e-3) - 1.0));
    for (int k = 0; k <= RQS_K; ++k) {
        float u  = (k == 0 || k == RQS_K) ? cst : ud[v * (RQS_K - 1) + (k - 1)];
        float sp = fmaxf(u, 0.0f) + log1pf(expf(-fabsf(u)));   // stable softplus
        ws[OFF_DV + v * RQS_STR + k] = 1e-3f + sp;
    }
    for (int k = 0; k < RQS_K; ++k)
        ws[OFF_DL + v * RQS_STR + k] =
            ws[OFF_H + v * RQS_STR + k] / ws[OFF_W + v * RQS_STR + k];
}

// ---------------- main elementwise kernel ----------------------------------

__device__ __forceinline__ void rqs_eval(float xi, int v, const float* T,
                                         float& o, float& l) {
    const float lo = -5.0f, hi = 5.0f;
    bool inside = (xi >= lo) && (xi <= hi);
    float x = fminf(fmaxf(xi, lo), hi);

    const float* CW = T + OFF_CW + v * RQS_STR;
    // branchless binary search: largest b in [0, K-1] with x >= CW[b]
    int b = 0;
    if (x >= CW[16])     b = 16;          // b+16 <= 29 always
    if (x >= CW[b + 8])  b += 8;          // b+8  <= 24 always
    if (x >= CW[b + 4])  b += 4;          // b+4  <= 28 always
    { int t = b + 2; if (t <= RQS_K - 1 && x >= CW[t]) b = t; }
    { int t = b + 1; if (t <= RQS_K - 1 && x >= CW[t]) b = t; }

    int base = v * RQS_STR + b;
    float in_cw = T[OFF_CW + base];
    float in_w  = T[OFF_W  + base];
    float in_ch = T[OFF_CH + base];
    float in_h  = T[OFF_H  + base];
    float dlt   = T[OFF_DL + base];
    float dk    = T[OFF_DV + base];
    float dk1   = T[OFF_DV + base + 1];

    float th   = (x - in_cw) / in_w;
    float omt  = 1.0f - th;
    float th1  = th * omt;
    float th2  = th * th;
    float num  = in_h * (dlt * th2 + dk * th1);
    float den  = dlt + (dk + dk1 - 2.0f * dlt) * th1;
    float dinv = 1.0f / den;
    float so   = in_ch + num * dinv;
    float dnum = dlt * dlt * (dk1 * th2 + 2.0f * dlt * th1 + dk * omt * omt);
    float sl   = logf(dnum * dinv * dinv);  // log(dnum) - 2*log(den)

    o = inside ? so : xi;
    l = inside ? sl : 0.0f;
}

__global__ __launch_bounds__(256) void rqs_main_kernel(
        const float* __restrict__ x,
        const float* __restrict__ tbl_g,
        float* __restrict__ out, int n) {
    __shared__ __align__(16) float T[TBL_FLOATS];

    // ---- async stage the 12.7KB table into LDS (gfx1250 async path) ----
    for (int i = threadIdx.x; i < TBL_VEC4; i += blockDim.x) {
        unsigned lds_byte = (unsigned)(unsigned long long)(&T[i * 4]); // LDS offset = low 32 bits
        unsigned goff     = (unsigned)(i * 16);
        asm volatile("global_load_async_to_lds_b128 %0, %1, %2"
                     :: "v"(lds_byte), "v"(goff), "s"(tbl_g)
                     : "memory");
    }
#if __has_builtin(__builtin_amdgcn_s_wait_asynccnt)
    __builtin_amdgcn_s_wait_asynccnt(0);
#else
    asm volatile("s_wait_asynccnt 0" ::: "memory");
#endif
    __syncthreads();

    // ---- streaming main loop: one float4 (4 consecutive v) per thread ----
    int n4 = n >> 2;
    int stride = gridDim.x * blockDim.x;
    const float4* xv = (const float4*)x;
    float4* ov = (float4*)out;
    float4* lv = (float4*)(out + n);

    for (int i = blockIdx.x * blockDim.x + threadIdx.x; i < n4; i += stride) {
        if (i + stride < n4)
            __builtin_prefetch((const void*)(xv + i + stride), 0, 3);  // global_prefetch_b8
        float4 xin = xv[i];
        int vb = (i & 3) << 2;       // v of first lane-element: (4*i) % 16
        float4 o, l;
        rqs_eval(xin.x, vb + 0, T, o.x, l.x);
        rqs_eval(xin.y, vb + 1, T, o.y, l.y);
        rqs_eval(xin.z, vb + 2, T, o.z, l.z);
        rqs_eval(xin.w, vb + 3, T, o.w, l.w);
        ov[i] = o;
        lv[i] = l;
    }

    // ---- scalar tail (n % 4 != 0; not hit for B*V = 8M) ----
    for (int idx = (n4 << 2) + blockIdx.x * blockDim.x + threadIdx.x;
         idx < n; idx += stride) {
        float oo, ll;
        rqs_eval(x[idx], idx & (RQS_V - 1), T, oo, ll);
        out[idx]     = oo;
        out[n + idx] = ll;
    }
}

// ---------------- host launcher --------------------------------------------

extern "C" void kernel_launch(void* const* d_in, const int* in_sizes, int n_in,
                              void* d_out, int out_size, void* d_ws, size_t ws_size,
                              hipStream_t stream) {
    (void)n_in; (void)out_size; (void)ws_size;
    const float* x  = (const float*)d_in[0];   // (B, V) inputs
    const float* uw = (const float*)d_in[1];   // (V, K) unnormalized widths
    const float* uh = (const float*)d_in[2];   // (V, K) unnormalized heights
    const float* ud = (const float*)d_in[3];   // (V, K-1) unnormalized derivatives
    float* out = (float*)d_out;                // [outputs(B*V) | logabsdet(B*V)]
    float* ws  = (float*)d_ws;                 // spline tables (12672 bytes used)
    int n = in_sizes[0];                       // B*V

    rqs_tables_kernel<<<1, 32, 0, stream>>>(uw, uh, ud, ws);

    int n4 = n >> 2;
    int blocks = 2048;
    int maxb = (n4 + 255) / 256;
    if (maxb < 1) maxb = 1;
    if (blocks > maxb) blocks = maxb;
    rqs_main_kernel<<<blocks, 256, 0, stream>>>(x, ws, out, n);
}